// Extraction_spatial_features_42975442764370
// MI455X (gfx1250) — compile-verified
//
#include <hip/hip_runtime.h>
#include <hip/hip_bf16.h>

// Problem constants (match reference setup_inputs)
#define BB   8
#define SS   24
#define NN   4096
#define KK   9
#define C1N  32
#define C2N  16
#define M2   (C1N * KK)   // 288
#define M3   (C2N * KK)   // 144
#define KN   (KK * NN)    // 36864
#define BNCNT (BB * NN)   // 32768 samples per (s,c) for BN

typedef float v2f __attribute__((ext_vector_type(2)));
typedef float v8f __attribute__((ext_vector_type(8)));

// Workspace layout (floats). Total = 37,750,272 floats ~ 151 MB.
constexpr size_t V1_OFF    = 0;                                   // (B,S,32,N)
constexpr size_t Y2_OFF    = (size_t)BB * SS * C1N * NN;          // (B,S,16,N)
constexpr size_t SUM_OFF   = Y2_OFF + (size_t)BB * SS * C2N * NN; // 384
constexpr size_t SQ_OFF    = SUM_OFF + 384;
constexpr size_t SCALE_OFF = SQ_OFF + 384;
constexpr size_t SHIFT_OFF = SCALE_OFF + 384;

// ---------------------------------------------------------------- zero stats
__global__ __launch_bounds__(256) void zero_stats_kernel(float* __restrict__ sum,
                                                         float* __restrict__ sq) {
  int i = threadIdx.x;
  for (; i < 384; i += 256) { sum[i] = 0.f; sq[i] = 0.f; }
}

// ---------------------------------------------------------------- stage 1
// v1[bs, o, n] = relu( b1[o] + sum_m W1[o,m] * x[bs, (n*9+m)%4096, (n*9+m)/4096] )
__global__ __launch_bounds__(256) void stage1_kernel(const float* __restrict__ x,
                                                     const float* __restrict__ W1,
                                                     const float* __restrict__ b1,
                                                     float* __restrict__ v1) {
  __shared__ float shW[C1N * KK + C1N];
  const int tid = threadIdx.x;
  for (int i = tid; i < C1N * KK; i += 256) shW[i] = W1[i];
  if (tid < C1N) shW[C1N * KK + tid] = b1[tid];
  __syncthreads();

  const int bs = blockIdx.y;
  const int n  = blockIdx.x * 256 + tid;
  const float* xb = x + (size_t)bs * NN * KK;

  float xv[KK];
#pragma unroll
  for (int m = 0; m < KK; ++m) {
    const int j  = n * KK + m;          // j < 36864
    const int ks = j >> 12;             // j / 4096
    const int ns = j & (NN - 1);        // j % 4096
    xv[m] = xb[ns * KK + ks];
  }
  float* vout = v1 + (size_t)bs * C1N * NN + n;   // stores coalesced in n
  for (int o = 0; o < C1N; ++o) {
    float acc = shW[C1N * KK + o];
#pragma unroll
    for (int m = 0; m < KK; ++m) acc = fmaf(shW[o * KK + m], xv[m], acc);
    vout[(size_t)o * NN] = fmaxf(acc, 0.f);
  }
}

// ---------------------------------------------------------------- stage 2 (WMMA fp32)
// y2[bs, o, n] = b2[o] + sum_{m<288} W2[o,m] * v1[bs, c(j), neigh[n'(j), k(j)]],
// j = n*288+m ; c = j/36864 (constant per 16-row tile), k = (j/4096)%9, n' = j&4095.
// One wave computes one 16x16 tile via 72 x V_WMMA_F32_16X16X4_F32,
// split over two independent accumulators to break the WMMA RAW chain.
__global__ __launch_bounds__(64) void stage2_wmma_kernel(const float* __restrict__ v1,
                                                         const int*   __restrict__ neigh,
                                                         const float* __restrict__ W2,
                                                         const float* __restrict__ b2,
                                                         float* __restrict__ y2,
                                                         float* __restrict__ sumbuf,
                                                         float* __restrict__ sqbuf) {
  __shared__ float shW2[C2N * M2];      // 18 KB, W2 row-major (16 x 288)
  __shared__ float shA[2][16 * 292];    // per-wave 16x288 A tile, padded stride

  const int tid  = threadIdx.x;
  const int wave = tid >> 5;
  const int lane = tid & 31;

  for (int i = tid; i < C2N * M2; i += 64) shW2[i] = W2[i];
  __syncthreads();

  const int tile = blockIdx.x * 2 + wave;     // 49152 tiles total
  const int bs   = tile >> 8;                 // 256 tiles per (b,s)
  const int n0   = (tile & 255) << 4;
  const int s    = bs % SS;
  const float* vb = v1 + (size_t)bs * C1N * NN;
  const int c0 = n0 >> 7;                     // single c channel for this tile

  // Prefetch the hot 16KB v1 channel row this tile gathers from (global_prefetch_b8)
  __builtin_prefetch((const char*)(vb + (size_t)c0 * NN) + lane * 512, 0, 3);

  // Stage A tile (16 rows x 288 cols) into LDS via double-indirect gather
  float* A = shA[wave];
  const int jbase = n0 * M2;
  for (int t = lane; t < 16 * M2; t += 32) {
    const int r  = t / M2;
    const int m  = t - r * M2;
    const int j  = jbase + t;                 // = (n0+r)*288 + m
    const int k  = (j >> 12) % KK;
    const int n2 = j & (NN - 1);
    const int nid = neigh[n2 * KK + k];
    A[r * 292 + m] = vb[(size_t)c0 * NN + nid];
  }
  // single wave produced and consumes its own LDS tile: DS ops are in-order per wave

  v8f acc0 = {};
  v8f acc1 = {};
  const int row  = lane & 15;                 // M (A) / N (B) index
  const int koff = (lane >> 4) << 1;          // 0 for lanes 0-15, 2 for 16-31
  for (int kk = 0; kk < M2 / 4; kk += 2) {    // 72 WMMA k-steps, 2 per iter
    const int k0 = kk * 4 + koff;
    v2f a0 = *(const v2f*)(A    + row * 292 + k0);
    v2f b0 = *(const v2f*)(shW2 + row * M2  + k0);
    v2f a1 = *(const v2f*)(A    + row * 292 + k0 + 4);
    v2f b1 = *(const v2f*)(shW2 + row * M2  + k0 + 4);
    acc0 = __builtin_amdgcn_wmma_f32_16x16x4_f32(false, a0, false, b0,
                                                 (short)0, acc0, false, false);
    acc1 = __builtin_amdgcn_wmma_f32_16x16x4_f32(false, a1, false, b1,
                                                 (short)0, acc1, false, false);
  }

  // bias + BN partial sums (each lane owns 8 rows of a single output channel o=row)
  const float bias = b2[row];
  float ls = 0.f, lsq = 0.f;
  float vals[8];
#pragma unroll
  for (int v = 0; v < 8; ++v) {
    const float val = (acc0[v] + acc1[v]) + bias;
    vals[v] = val;
    ls += val;
    lsq = fmaf(val, val, lsq);
  }

  // transpose tile through LDS to get y2 stores contiguous along n
  const int half = lane >> 4;
#pragma unroll
  for (int v = 0; v < 8; ++v) A[row * 16 + v + half * 8] = vals[v];  // T[o*16 + r]

  const int o  = lane >> 1;
  const int rb = (lane & 1) << 3;
  const float4 t0 = *(const float4*)(A + o * 16 + rb);
  const float4 t1 = *(const float4*)(A + o * 16 + rb + 4);
  float* dst = y2 + ((size_t)bs * C2N + o) * NN + n0 + rb;
  *(float4*)dst       = t0;
  *(float4*)(dst + 4) = t1;

  // fold the two lane-halves (rows 0-7 and 8-15 of channel o) and atomically accumulate
  ls  += __shfl_xor(ls, 16, 32);
  lsq += __shfl_xor(lsq, 16, 32);
  if (lane < 16) {
    atomicAdd(&sumbuf[s * C2N + row], ls);
    atomicAdd(&sqbuf [s * C2N + row], lsq);
  }
}

// ---------------------------------------------------------------- BN finalize
__global__ __launch_bounds__(384) void finalize_stats_kernel(const float* __restrict__ sum,
                                                             const float* __restrict__ sq,
                                                             const float* __restrict__ gamma,
                                                             const float* __restrict__ beta,
                                                             float* __restrict__ scale,
                                                             float* __restrict__ shift) {
  const int idx = threadIdx.x;
  if (idx >= SS * C2N) return;
  const int c = idx & (C2N - 1);
  const float inv_n = 1.0f / (float)BNCNT;
  const float mean = sum[idx] * inv_n;
  const float var  = fmaf(-mean, mean, sq[idx] * inv_n);
  const float sc   = gamma[c] * rsqrtf(var + 1e-5f);
  scale[idx] = sc;
  shift[idx] = fmaf(-mean, sc, beta[c]);
}

// ---------------------------------------------------------------- stage 3
// out[bs,n] = relu( b3 + sum_{m<144} W3[m] * relu(scale*y2[bs,c,neigh[n',k]]+shift) )
__global__ __launch_bounds__(256) void stage3_kernel(const float* __restrict__ y2,
                                                     const int*   __restrict__ neigh,
                                                     const float* __restrict__ W3,
                                                     const float* __restrict__ b3,
                                                     const float* __restrict__ scale,
                                                     const float* __restrict__ shift,
                                                     float* __restrict__ out) {
  __shared__ float shW3[M3];
  __shared__ float shSc[C2N];
  __shared__ float shSh[C2N];
  const int tid = threadIdx.x;
  const int bs  = blockIdx.y;
  const int s   = bs % SS;
  if (tid < M3)  shW3[tid] = W3[tid];
  if (tid < C2N) { shSc[tid] = scale[s * C2N + tid]; shSh[tid] = shift[s * C2N + tid]; }
  __syncthreads();

  const int n = blockIdx.x * 256 + tid;
  const float* yb = y2 + (size_t)bs * C2N * NN;
  const int c = n >> 8;          // j spans never cross a 36864 boundary within one n
  const float sc = shSc[c], sh = shSh[c];
  const float* yc = yb + (size_t)c * NN;

  float acc = b3[0];
  const int j0 = n * M3;
#pragma unroll 4
  for (int m = 0; m < M3; ++m) {
    const int j  = j0 + m;
    const int k  = (j >> 12) % KK;
    const int n2 = j & (NN - 1);
    const int nid = neigh[n2 * KK + k];
    const float h = fmaxf(fmaf(yc[nid], sc, sh), 0.f);
    acc = fmaf(shW3[m], h, acc);
  }
  out[(size_t)bs * NN + n] = fmaxf(acc, 0.f);
}

// ---------------------------------------------------------------- launch
extern "C" void kernel_launch(void* const* d_in, const int* in_sizes, int n_in,
                              void* d_out, int out_size, void* d_ws, size_t ws_size,
                              hipStream_t stream) {
  const float* x     = (const float*)d_in[0];
  const int*   neigh = (const int*)  d_in[1];
  const float* W1    = (const float*)d_in[2];
  const float* b1    = (const float*)d_in[3];
  const float* W2    = (const float*)d_in[4];
  const float* b2    = (const float*)d_in[5];
  const float* gamma = (const float*)d_in[6];
  const float* beta  = (const float*)d_in[7];
  const float* W3    = (const float*)d_in[8];
  const float* b3    = (const float*)d_in[9];
  float* out = (float*)d_out;
  float* ws  = (float*)d_ws;   // needs >= ~151 MB (see offsets above)

  float* v1    = ws + V1_OFF;
  float* y2    = ws + Y2_OFF;
  float* sum   = ws + SUM_OFF;
  float* sq    = ws + SQ_OFF;
  float* scale = ws + SCALE_OFF;
  float* shift = ws + SHIFT_OFF;

  zero_stats_kernel<<<1, 256, 0, stream>>>(sum, sq);
  stage1_kernel<<<dim3(NN / 256, BB * SS), 256, 0, stream>>>(x, W1, b1, v1);
  // 49152 tiles, 2 waves (tiles) per 64-thread block
  stage2_wmma_kernel<<<(BB * SS * (NN / 16)) / 2, 64, 0, stream>>>(v1, neigh, W2, b2,
                                                                   y2, sum, sq);
  finalize_stats_kernel<<<1, 384, 0, stream>>>(sum, sq, gamma, beta, scale, shift);
  stage3_kernel<<<dim3(NN / 256, BB * SS), 256, 0, stream>>>(y2, neigh, W3, b3,
                                                             scale, shift, out);
}